// AdaptivePatchSequenceBuilder_11115375362922
// MI455X (gfx1250) — compile-verified
//
#include <hip/hip_runtime.h>
#include <hip/hip_bf16.h>
#include <stddef.h>

// ---------------------------------------------------------------------------
// AdaptivePatchSequenceBuilder on MI455X (gfx1250), fp32 WMMA path.
// Projection GEMMs: LDS-tiled (32Mx64N per 256-thread block, K chunks of 128),
// fragments fed from LDS (ds_load_b64), WMMA = V_WMMA_F32_16X16X4_F32.
// ---------------------------------------------------------------------------

typedef float v2f __attribute__((ext_vector_type(2)));
typedef float v4f __attribute__((ext_vector_type(4)));
typedef float v8f __attribute__((ext_vector_type(8)));

#define DMODEL 768
#define IMG 2048
#define GRID 128
#define GRID2 64
#define LPITCH 132   // LDS row pitch in floats; 132 % 64 == 4 -> conflict-free frags

// ---------------------------------------------------------------------------
// 1) Mask compaction: ordered index lists + counts (single 1024-thread block)
// ---------------------------------------------------------------------------
__global__ __launch_bounds__(1024) void scan_masks_kernel(
    const unsigned char* __restrict__ m16, const unsigned char* __restrict__ m32,
    int* __restrict__ nbuf, int* __restrict__ idx16, int* __restrict__ idx32)
{
  __shared__ int s[1024];
  const int tid = threadIdx.x;

  int cnt = 0;
  for (int i = 0; i < 16; ++i) cnt += (m16[tid * 16 + i] != 0);
  s[tid] = cnt;
  __syncthreads();
  for (int off = 1; off < 1024; off <<= 1) {
    int t = (tid >= off) ? s[tid - off] : 0;
    __syncthreads();
    s[tid] += t;
    __syncthreads();
  }
  int pos = s[tid] - cnt;
  for (int i = 0; i < 16; ++i) {
    int ii = tid * 16 + i;
    if (m16[ii]) idx16[pos++] = ii;
  }
  if (tid == 1023) nbuf[0] = s[1023];
  __syncthreads();

  cnt = 0;
  for (int i = 0; i < 4; ++i) cnt += (m32[tid * 4 + i] != 0);
  s[tid] = cnt;
  __syncthreads();
  for (int off = 1; off < 1024; off <<= 1) {
    int t = (tid >= off) ? s[tid - off] : 0;
    __syncthreads();
    s[tid] += t;
    __syncthreads();
  }
  pos = s[tid] - cnt;
  for (int i = 0; i < 4; ++i) {
    int ii = tid * 4 + i;
    if (m32[ii]) idx32[pos++] = ii;
  }
  if (tid == 1023) nbuf[1] = s[1023];
}

// ---------------------------------------------------------------------------
// 2) cls token + its positional embedding -> output row 0
// ---------------------------------------------------------------------------
__global__ void cls_kernel(const float* __restrict__ cls_tok,
                           const float* __restrict__ pos,
                           float* __restrict__ out)
{
  int d = blockIdx.x * blockDim.x + threadIdx.x;
  if (d < DMODEL) out[d] = cls_tok[d] + pos[d];
}

// ---------------------------------------------------------------------------
// 3) bilinear x0.5 (align_corners=False, no antialias) == exact 2x2 average
// ---------------------------------------------------------------------------
__global__ void downscale_kernel(const float* __restrict__ img,
                                 float* __restrict__ imgh)
{
  int idx = blockIdx.x * blockDim.x + threadIdx.x;   // over 3*1024*1024
  if (idx >= 3 * 1024 * 1024) return;
  int c = idx >> 20;
  int rem = idx & ((1 << 20) - 1);
  int y = rem >> 10, x = rem & 1023;
  const float* p = img + ((size_t)c * IMG + 2 * y) * IMG + 2 * x;
  imgh[idx] = 0.25f * (p[0] + p[1] + p[IMG] + p[IMG + 1]);
}

// ---------------------------------------------------------------------------
// 4/5) separable bicubic (Keys a=-0.5) half-scale with antialias:
// 8 taps at offsets d = -3..4 around 2*i, u = 0.5*(d-0.5), w = 0.5*cubic(u).
// ---------------------------------------------------------------------------
__device__ __constant__ float kW8[8] = {
    -0.01171875f, -0.03515625f, 0.11328125f, 0.43359375f,
     0.43359375f,  0.11328125f, -0.03515625f, -0.01171875f};

__global__ void pos_rows_kernel(const float* __restrict__ pg,  // (128,128,768)
                                float* __restrict__ tmp)       // (64,128,768)
{
  int idx = blockIdx.x * blockDim.x + threadIdx.x;   // 64*128*768
  if (idx >= GRID2 * GRID * DMODEL) return;
  int d = idx % DMODEL;
  int t = idx / DMODEL;
  int x = t & (GRID - 1);
  int y2 = t >> 7;
  float acc = 0.f, wsum = 0.f;
  #pragma unroll
  for (int tp = 0; tp < 8; ++tp) {
    int j = 2 * y2 - 3 + tp;
    if (j >= 0 && j < GRID) {
      acc += kW8[tp] * pg[((size_t)j * GRID + x) * DMODEL + d];
      wsum += kW8[tp];
    }
  }
  tmp[idx] = acc / wsum;
}

__global__ void pos_cols_kernel(const float* __restrict__ tmp,  // (64,128,768)
                                float* __restrict__ pg2)        // (64,64,768)
{
  int idx = blockIdx.x * blockDim.x + threadIdx.x;   // 64*64*768
  if (idx >= GRID2 * GRID2 * DMODEL) return;
  int d = idx % DMODEL;
  int t = idx / DMODEL;
  int x2 = t & (GRID2 - 1);
  int y2 = t >> 6;
  float acc = 0.f, wsum = 0.f;
  #pragma unroll
  for (int tp = 0; tp < 8; ++tp) {
    int j = 2 * x2 - 3 + tp;
    if (j >= 0 && j < GRID) {
      acc += kW8[tp] * tmp[((size_t)y2 * GRID + j) * DMODEL + d];
      wsum += kW8[tp];
    }
  }
  pg2[idx] = acc / wsum;
}

// ---------------------------------------------------------------------------
// 6-8) LDS-tiled projection GEMM: [rows, 768] x proj_w^T, 16x16x4 fp32 WMMA.
//   Block = 256 threads = 8 waves arranged 2(M) x 4(N): 32 rows x 64 cols.
//   K loop in chunks of 128: stage A(32x128) + B(64x128) in LDS (coalesced
//   global_load_b128), then 32 WMMA steps per wave fed by ds_load_b64.
//   Fragment layout (wave32): lhalf=lane>>4, lm=lane&15:
//     A row M=lm, K=kk+2*lhalf(+1); B col N=lm, same K; D elem v -> (v+8*lhalf, lm)
// MODE 0: tok16 rows from image patches; epilogue +proj_b +pos[1+g]
// MODE 1: tok32 rows from half image;    epilogue +proj_b +pg2[g]
// MODE 2: fine tokens (4 per group);     epilogue +proj_b -> ft buffer
// ---------------------------------------------------------------------------
template <int MODE>
__global__ __launch_bounds__(256) void gemm_proj_kernel(
    const float* __restrict__ src, const float* __restrict__ proj_w,
    const float* __restrict__ proj_b, const float* __restrict__ aux,
    const int* __restrict__ nbuf, const int* __restrict__ idx,
    float* __restrict__ dst)
{
  const int n1 = nbuf[0], n2 = nbuf[1];
  const int rows = (MODE == 0) ? n1 : (MODE == 1 ? n2 : 4 * n2);
  const int blockM = blockIdx.x;            // 32 rows
  const int tileN0 = blockIdx.y * 64;       // 64 cols
  if (blockM * 32 >= rows) return;          // uniform exit; rows >= 1 past here

  __shared__ float As[32][LPITCH];
  __shared__ float Bs[64][LPITCH];

  const int tid = threadIdx.x;

  // --- A-staging assignment: thread t -> row (t>>3), 16-float segment (t&7) ---
  const int mloc = tid >> 3;
  int rA = blockM * 32 + mloc;
  int rAc = (rA < rows) ? rA : (rows - 1);
  const float* pimg;
  int lineStride;
  size_t cStride;
  if (MODE == 1) {
    int g = idx[rAc];
    pimg = src + (size_t)((g >> 6) * 16) * 1024 + (g & 63) * 16;
    lineStride = 1024; cStride = (size_t)1024 * 1024;
  } else if (MODE == 0) {
    int g = idx[rAc];
    pimg = src + (size_t)((g >> 7) * 16) * IMG + (g & 127) * 16;
    lineStride = IMG; cStride = (size_t)IMG * IMG;
  } else {
    int g = idx[rAc >> 2];
    int sub = rAc & 3;
    int by = (g >> 6) * 2 + (sub >> 1);
    int bx = (g & 63) * 2 + (sub & 1);
    pimg = src + (size_t)(by * 16) * IMG + bx * 16;
    lineStride = IMG; cStride = (size_t)IMG * IMG;
  }

  // --- B-staging assignment: thread t -> weight row (t>>2), 32-float seg (t&3) ---
  const int nloc = tid >> 2;
  const float* pwrow = proj_w + (size_t)(tileN0 + nloc) * DMODEL;

  const int wave = tid >> 5;
  const int lane = tid & 31;
  const int lhalf = lane >> 4, lm = lane & 15;
  const int wM = (wave >> 2) * 16;   // 0 or 16
  const int wN = (wave & 3) * 16;    // 0,16,32,48

  v8f acc = {};
  for (int kc = 0; kc < DMODEL; kc += 128) {
    // stage A chunk: each 16-k segment is one 64B-contiguous pixel row
    {
      int ks = (tid & 7) * 16;
      int k = kc + ks;
      int c = k >> 8, rem = k & 255, h = rem >> 4;
      const float* p = pimg + cStride * c + (size_t)h * lineStride;
      v4f x0 = *(const v4f*)(p);
      v4f x1 = *(const v4f*)(p + 4);
      v4f x2 = *(const v4f*)(p + 8);
      v4f x3 = *(const v4f*)(p + 12);
      *(v4f*)&As[mloc][ks]      = x0;
      *(v4f*)&As[mloc][ks + 4]  = x1;
      *(v4f*)&As[mloc][ks + 8]  = x2;
      *(v4f*)&As[mloc][ks + 12] = x3;
    }
    // stage B chunk: weight rows are fully contiguous in proj_w
    {
      int ks = (tid & 3) * 32;
      const float* p = pwrow + kc + ks;
      #pragma unroll
      for (int q = 0; q < 8; ++q)
        *(v4f*)&Bs[nloc][ks + 4 * q] = *(const v4f*)(p + 4 * q);
    }
    __syncthreads();
    #pragma unroll 8
    for (int kk = 0; kk < 128; kk += 4) {
      int k2 = kk + 2 * lhalf;
      v2f a = *(const v2f*)&As[wM + lm][k2];
      v2f b = *(const v2f*)&Bs[wN + lm][k2];
      acc = __builtin_amdgcn_wmma_f32_16x16x4_f32(false, a, false, b,
                                                  (short)0, acc, false, false);
    }
    __syncthreads();
  }

  // epilogue
  #pragma unroll
  for (int v = 0; v < 8; ++v) {
    int m = blockM * 32 + wM + v + 8 * lhalf;
    if (m < rows) {
      int n = tileN0 + wN + lm;
      if (MODE == 0) {
        int gm = idx[m];
        dst[(size_t)(1 + m) * DMODEL + n] =
            acc[v] + proj_b[n] + aux[(size_t)(1 + gm) * DMODEL + n];
      } else if (MODE == 1) {
        int gm = idx[m];
        dst[(size_t)(1 + n1 + m) * DMODEL + n] =
            acc[v] + proj_b[n] + aux[(size_t)gm * DMODEL + n];
      } else {
        dst[(size_t)m * DMODEL + n] = acc[v] + proj_b[n];
      }
    }
  }
}

// ---------------------------------------------------------------------------
// 9) attn : [N2,3072] x attn_w^T + attn_b ; A[n, k=(c*4+s)] = ft[(4n+s)*768+c]
//    (per-wave 16x16 tiles; A/B float2 loads straight from L2)
// ---------------------------------------------------------------------------
__global__ __launch_bounds__(128) void gemm_attn_kernel(
    const float* __restrict__ ft, const float* __restrict__ attn_w,
    const float* __restrict__ attn_b, const int* __restrict__ nbuf,
    float* __restrict__ attn)
{
  const int n2 = nbuf[1];
  const int tileM = blockIdx.x * 4 + (threadIdx.x >> 5);
  const int tileN = blockIdx.y;
  if (tileM * 16 >= n2) return;
  const int lane = threadIdx.x & 31;
  const int lhalf = lane >> 4, lm = lane & 15;

  int r = tileM * 16 + lm;
  int rc = (r < n2) ? r : (n2 - 1);
  const float* fa = ft + (size_t)rc * 4 * DMODEL;
  const float* pw = attn_w + (size_t)(tileN * 16 + lm) * (4 * DMODEL);

  v8f acc = {};
  for (int kb = 0; kb < 4 * DMODEL; kb += 4) {
    int k = kb + 2 * lhalf;              // k even -> s in {0,2}; k+1 same c
    int c = k >> 2, s = k & 3;
    const float* p = fa + (size_t)s * DMODEL + c;
    v2f a;
    a.x = p[0];
    a.y = p[DMODEL];
    v2f b = *(const v2f*)(pw + k);
    acc = __builtin_amdgcn_wmma_f32_16x16x4_f32(false, a, false, b,
                                                (short)0, acc, false, false);
  }
  #pragma unroll
  for (int v = 0; v < 8; ++v) {
    int m = tileM * 16 + v + 8 * lhalf;
    if (m < n2) {
      int n = tileN * 16 + lm;
      attn[(size_t)m * DMODEL + n] = acc[v] + attn_b[n];
    }
  }
}

// ---------------------------------------------------------------------------
// 10) zero-conv : out_rows(tok32) += attn x zc_w^T + zc_b
// ---------------------------------------------------------------------------
__global__ __launch_bounds__(128) void gemm_zc_kernel(
    const float* __restrict__ attn, const float* __restrict__ zc_w,
    const float* __restrict__ zc_b, const int* __restrict__ nbuf,
    float* __restrict__ out)
{
  const int n1 = nbuf[0], n2 = nbuf[1];
  const int tileM = blockIdx.x * 4 + (threadIdx.x >> 5);
  const int tileN = blockIdx.y;
  if (tileM * 16 >= n2) return;
  const int lane = threadIdx.x & 31;
  const int lhalf = lane >> 4, lm = lane & 15;

  int r = tileM * 16 + lm;
  int rc = (r < n2) ? r : (n2 - 1);
  const float* pa = attn + (size_t)rc * DMODEL;
  const float* pw = zc_w + (size_t)(tileN * 16 + lm) * DMODEL;

  v8f acc = {};
  for (int kb = 0; kb < DMODEL; kb += 4) {
    int k = kb + 2 * lhalf;
    v2f a = *(const v2f*)(pa + k);
    v2f b = *(const v2f*)(pw + k);
    acc = __builtin_amdgcn_wmma_f32_16x16x4_f32(false, a, false, b,
                                                (short)0, acc, false, false);
  }
  #pragma unroll
  for (int v = 0; v < 8; ++v) {
    int m = tileM * 16 + v + 8 * lhalf;
    if (m < n2) {
      int n = tileN * 16 + lm;
      float* p = out + (size_t)(1 + n1 + m) * DMODEL + n;
      *p = *p + acc[v] + zc_b[n];
    }
  }
}

// ---------------------------------------------------------------------------
// 11) second return value: selected_patch_count = N1 + N2
// ---------------------------------------------------------------------------
__global__ void count_kernel(const int* __restrict__ nbuf,
                             float* __restrict__ out, int out_size)
{
  if (threadIdx.x == 0 && blockIdx.x == 0)
    out[out_size - 1] = (float)(nbuf[0] + nbuf[1]);
}

// ---------------------------------------------------------------------------
extern "C" void kernel_launch(void* const* d_in, const int* in_sizes, int n_in,
                              void* d_out, int out_size, void* d_ws, size_t ws_size,
                              hipStream_t stream) {
  (void)in_sizes; (void)n_in; (void)ws_size;
  const float* image   = (const float*)d_in[0];
  const unsigned char* mask16 = (const unsigned char*)d_in[1];
  const unsigned char* mask32 = (const unsigned char*)d_in[2];
  const float* proj_w  = (const float*)d_in[3];
  const float* proj_b  = (const float*)d_in[4];
  const float* cls_tok = (const float*)d_in[5];
  const float* pos     = (const float*)d_in[6];   // (1+128*128, 768)
  const float* attn_w  = (const float*)d_in[7];
  const float* attn_b  = (const float*)d_in[8];
  const float* zc_w    = (const float*)d_in[9];
  const float* zc_b    = (const float*)d_in[10];
  float* out = (float*)d_out;

  // workspace layout (~85 MB)
  char* w = (char*)d_ws;
  int*   nbuf   = (int*)(w + 0);                    // [0]=N1 [1]=N2
  int*   idx16  = (int*)(w + 256);                  // 16384 ints
  int*   idx32  = (int*)(w + 256 + 65536);          // 4096 ints
  size_t off    = 82176;                            // 256-aligned
  float* imgh   = (float*)(w + off);  off += (size_t)3 * 1024 * 1024 * 4;
  float* pg2    = (float*)(w + off);  off += (size_t)GRID2 * GRID2 * DMODEL * 4;
  float* big    = (float*)(w + off);  off += (size_t)16384 * DMODEL * 4;
  float* attn   = (float*)(w + off);  off += (size_t)4096 * DMODEL * 4;
  float* tmpP   = big;   // (64,128,768) pos-resize temp, dead before ft written
  float* ft     = big;   // (4*N2max, 768)

  scan_masks_kernel<<<1, 1024, 0, stream>>>(mask16, mask32, nbuf, idx16, idx32);
  cls_kernel<<<3, 256, 0, stream>>>(cls_tok, pos, out);
  downscale_kernel<<<(3 * 1024 * 1024 + 255) / 256, 256, 0, stream>>>(image, imgh);
  pos_rows_kernel<<<(GRID2 * GRID * DMODEL + 255) / 256, 256, 0, stream>>>(pos + DMODEL, tmpP);
  pos_cols_kernel<<<(GRID2 * GRID2 * DMODEL + 255) / 256, 256, 0, stream>>>(tmpP, pg2);

  // tiled projection GEMMs: grid.x covers worst-case rows, blocks self-trim
  gemm_proj_kernel<0><<<dim3(512, DMODEL / 64), 256, 0, stream>>>(
      image, proj_w, proj_b, pos, nbuf, idx16, out);
  gemm_proj_kernel<1><<<dim3(128, DMODEL / 64), 256, 0, stream>>>(
      imgh, proj_w, proj_b, pg2, nbuf, idx32, out);
  gemm_proj_kernel<2><<<dim3(512, DMODEL / 64), 256, 0, stream>>>(
      image, proj_w, proj_b, nullptr, nbuf, idx32, ft);

  gemm_attn_kernel<<<dim3(64, DMODEL / 16), 128, 0, stream>>>(
      ft, attn_w, attn_b, nbuf, attn);
  gemm_zc_kernel<<<dim3(64, DMODEL / 16), 128, 0, stream>>>(
      attn, zc_w, zc_b, nbuf, out);
  count_kernel<<<1, 1, 0, stream>>>(nbuf, out, out_size);
}